// PartialCELoss_72103910965336
// MI455X (gfx1250) — compile-verified
//
#include <hip/hip_runtime.h>
#include <stdint.h>

// Partial-label CE loss, MI455X (gfx1250).
// Bandwidth-bound: 524MB streamed once -> ~22.5us floor at 23.3 TB/s.
// Strategy: one wave32 per row, single-pass online softmax, async
// global->LDS triple-buffered streaming (ASYNCcnt, 2 tiles in flight,
// non-temporal hint: read-once stream > 192MB L2), deterministic two-stage
// reduction (no float atomics).

#define CLS   1000            // classes per row (fixed by reference)
#define TILE  128             // floats per tile per array (512B / wave)
#define NTILE ((CLS + TILE - 1) / TILE)   // 8
#define NBUF  3               // triple buffer: 2 tiles in flight
#define WAVES 8               // waves per block (256 threads, wave32)
#define EPSV  1e-5f

// ---- CDNA5 async tensor-path ops (inline asm; ASYNCcnt-tracked) ----
__device__ __forceinline__ void async_ld_b128(uint32_t lds_off, uint64_t gaddr) {
  // VDST = per-lane LDS byte offset, VADDR = per-lane 64-bit global address.
  // NT: read-once stream, don't pollute L2.
  asm volatile("global_load_async_to_lds_b128 %0, %1, off th:TH_LOAD_NT"
               :: "v"(lds_off), "v"(gaddr) : "memory");
}
template <int N>
__device__ __forceinline__ void wait_async() {
  if constexpr (N >= 4)      asm volatile("s_wait_asynccnt 4" ::: "memory");
  else if constexpr (N >= 2) asm volatile("s_wait_asynccnt 2" ::: "memory");
  else                       asm volatile("s_wait_asynccnt 0" ::: "memory");
}

__global__ __launch_bounds__(256)
void pce_rows(const float* __restrict__ logits, const float* __restrict__ Y,
              float* __restrict__ partial, int B) {
  // [wave][triple-buffer][logits|Y][128 floats]  = 24KB
  __shared__ __align__(16) float sh[WAVES][NBUF][2][TILE];
  __shared__ float stT[WAVES], stC[WAVES];

  const int wave = threadIdx.x >> 5;
  const int lane = threadIdx.x & 31;
  const int row  = blockIdx.x * WAVES + wave;

  // online-softmax state (per lane)
  float m = -3.0e38f, se = 0.f, sy = 0.f, ny = 0.f;

  if (row < B) {
    const char* rowL = (const char*)(logits + (size_t)row * CLS);
    const char* rowY = (const char*)(Y      + (size_t)row * CLS);

    // LDS byte offsets for this lane's 16B slot in each buffer
    uint32_t ldsL[NBUF], ldsY[NBUF];
#pragma unroll
    for (int b = 0; b < NBUF; ++b) {
      ldsL[b] = (uint32_t)(uintptr_t)&sh[wave][b][0][lane * 4];
      ldsY[b] = (uint32_t)(uintptr_t)&sh[wave][b][1][lane * 4];
    }

    auto issue = [&](int t) {
      const int buf = t % NBUF;
      int off = t * (TILE * 4) + lane * 16;               // byte offset within row
      off = off > (CLS * 4 - 16) ? (CLS * 4 - 16) : off;  // clamp: stay in row
      async_ld_b128(ldsL[buf], (uint64_t)(uintptr_t)(rowL + off));
      async_ld_b128(ldsY[buf], (uint64_t)(uintptr_t)(rowY + off));
    };

    issue(0);
    issue(1);
#pragma unroll
    for (int t = 0; t < NTILE; ++t) {
      // keep 2 tiles in flight; waits guarantee tile t is complete
      if (t + 2 < NTILE)      { issue(t + 2); wait_async<4>(); }
      else if (t + 1 < NTILE) { wait_async<2>(); }
      else                    { wait_async<0>(); }
      const int buf = t % NBUF;
      const int idx = t * TILE + lane * 4;  // 1000 % 4 == 0: whole vec valid or not
      if (idx < CLS) {
        float4 x = *(const float4*)&sh[wave][buf][0][lane * 4];
        float4 y = *(const float4*)&sh[wave][buf][1][lane * 4];
        float mx = fmaxf(fmaxf(x.x, x.y), fmaxf(x.z, x.w));
        float nm = fmaxf(m, mx);
        float sc = __expf(m - nm);          // first iter: exp(-inf)=0, se/sy are 0
        float e0 = __expf(x.x - nm), e1 = __expf(x.y - nm);
        float e2 = __expf(x.z - nm), e3 = __expf(x.w - nm);
        se = se * sc + ((e0 + e1) + (e2 + e3));
        sy = sy * sc + ((y.x * e0 + y.y * e1) + (y.z * e2 + y.w * e3));
        ny += ((y.x + y.y) + (y.z + y.w));
        m = nm;
      }
    }
  }

  // wave32 butterfly merge of online-softmax state (MI455X is wave32)
#pragma unroll
  for (int o = 16; o >= 1; o >>= 1) {
    float m2  = __shfl_xor(m,  o, 32);
    float se2 = __shfl_xor(se, o, 32);
    float sy2 = __shfl_xor(sy, o, 32);
    float ny2 = __shfl_xor(ny, o, 32);
    float M = fmaxf(m, m2);
    float a = __expf(m - M), b = __expf(m2 - M);
    se = se * a + se2 * b;
    sy = sy * a + sy2 * b;
    ny += ny2;
    m = M;
  }

  if (lane == 0) {
    float loss = 0.f, msk = 0.f;
    if (row < B) {
      float s = sy / se;                              // sum(Y * softmax(row))
      msk  = (ny <= (float)(CLS - 1)) ? 1.f : 0.f;    // candidate set not full
      loss = -__logf(s + EPSV) * msk;
    }
    stT[wave] = loss;
    stC[wave] = msk;
  }
  __syncthreads();
  if (threadIdx.x == 0) {                             // fixed-order -> deterministic
    float T = 0.f, Cc = 0.f;
#pragma unroll
    for (int i = 0; i < WAVES; ++i) { T += stT[i]; Cc += stC[i]; }
    partial[2 * blockIdx.x]     = T;
    partial[2 * blockIdx.x + 1] = Cc;
  }
}

__global__ __launch_bounds__(256)
void pce_final(const float* __restrict__ partial, int n, float* __restrict__ out) {
  __shared__ float sT[256], sC[256];
  float T = 0.f, Cc = 0.f;
  for (int i = threadIdx.x; i < n; i += 256) {        // fixed strided order
    T  += partial[2 * i];
    Cc += partial[2 * i + 1];
  }
  sT[threadIdx.x] = T; sC[threadIdx.x] = Cc;
  __syncthreads();
  for (int s = 128; s > 0; s >>= 1) {                 // fixed tree order
    if (threadIdx.x < s) {
      sT[threadIdx.x] += sT[threadIdx.x + s];
      sC[threadIdx.x] += sC[threadIdx.x + s];
    }
    __syncthreads();
  }
  if (threadIdx.x == 0)
    out[0] = (sC[0] > 0.f) ? (sT[0] / sC[0]) : sT[0];
}

extern "C" void kernel_launch(void* const* d_in, const int* in_sizes, int n_in,
                              void* d_out, int out_size, void* d_ws, size_t ws_size,
                              hipStream_t stream) {
  const float* logits = (const float*)d_in[0];
  const float* Y      = (const float*)d_in[1];
  float* out  = (float*)d_out;
  float* part = (float*)d_ws;                 // 2 floats per block (64KB @ B=65536)

  const int B  = in_sizes[0] / CLS;           // 65536 for the reference shapes
  const int nb = (B + WAVES - 1) / WAVES;     // 8192 blocks

  pce_rows<<<nb, 256, 0, stream>>>(logits, Y, part, B);
  pce_final<<<1, 256, 0, stream>>>(part, nb, out);
}